// ResidualGatedGCNLayer_48473000903539
// MI455X (gfx1250) — compile-verified
//
#include <hip/hip_runtime.h>
#include <hip/hip_bf16.h>
#include <math.h>

#define B_ 8
#define N_ 256
#define H_ 128
#define EPS_ 1e-5f

typedef float v2f __attribute__((ext_vector_type(2)));
typedef float v8f __attribute__((ext_vector_type(8)));

// ---- workspace layout (float element offsets), total ~4.3 MB ----
#define WS_VX    0                       // Vx  [2048*128]
#define WS_VNX   262144                  // Vnx [2048*128]
#define WS_AGG   524288                  // agg [2048*128]
#define WS_XNEW  786432                  // x_new [2048*128]
#define WS_EPART 1048576                 // 64 slots * 256 (sum | sumsq)
#define WS_XPART (WS_EPART + 64*256)
#define WS_STATS (WS_XPART + 64*256)     // me[128] re[128] mx[128] rx[128]
#define WS_ZLEN  (WS_STATS + 512 - WS_AGG)

#define TILE_F   4224                    // 32 rows * 132 floats (skewed)

__global__ __launch_bounds__(256) void k_init(float* __restrict__ ws) {
  for (int idx = blockIdx.x * 256 + threadIdx.x; idx < WS_ZLEN; idx += gridDim.x * 256)
    ws[WS_AGG + idx] = 0.0f;
}

// Vx = x @ Ve_w^T + Ve_b ; Vnx = x @ Vn_w^T + Vn_b   (tiny: 2048x128x128)
__global__ __launch_bounds__(256) void k_vxvnx(const float* __restrict__ x,
    const float* __restrict__ Vew, const float* __restrict__ Veb,
    const float* __restrict__ Vnw, const float* __restrict__ Vnb,
    float* __restrict__ ws) {
  int idx = blockIdx.x * 256 + threadIdx.x;            // 262144 total
  int row = idx >> 7, h = idx & 127;
  const float* xr = x + row * H_;
  const float* wv = Vew + h * H_;
  const float* wn = Vnw + h * H_;
  float av = 0.f, an = 0.f;
#pragma unroll
  for (int k4 = 0; k4 < 32; ++k4) {
    float4 xv = *(const float4*)(xr + 4 * k4);
    float4 v  = *(const float4*)(wv + 4 * k4);
    float4 n  = *(const float4*)(wn + 4 * k4);
    av += xv.x * v.x + xv.y * v.y + xv.z * v.z + xv.w * v.w;
    an += xv.x * n.x + xv.y * n.y + xv.z * n.z + xv.w * n.w;
  }
  ws[WS_VX  + idx] = av + Veb[h];
  ws[WS_VNX + idx] = an + Vnb[h];
}

// Issue async global->LDS copy of one 32x128 f32 tile (skewed stride 132) into buffer `buf`.
// The single __shared__ object sits at LDS offset 0, so raw LDS offsets are computed directly.
__device__ __forceinline__ void issue_tile_async(const float* gbase, int tid, int buf) {
#pragma unroll
  for (int q = 0; q < 4; ++q) {
    int idx = tid + 256 * q;
    int row = idx >> 5, c4 = (idx & 31) << 2;
    unsigned long long ga = (unsigned long long)(size_t)(gbase + row * H_ + c4);
    unsigned int       lo = (unsigned int)((buf * TILE_F + row * 132 + c4) * 4);
    asm volatile("global_load_async_to_lds_b128 %0, %1, off"
                 :: "v"(lo), "v"(ga) : "memory");
  }
}

// Main e-path: one block per (b,i), loops over 8 j-tiles with double-buffered async LDS loads.
// PASS2=false: BN stats + gated aggregation (flash pass, no e_new stored).
// PASS2=true : recompute e_new, BN + relu + residual -> eout (residual read back from LDS tile).
template <bool PASS2>
__global__ __launch_bounds__(256) void k_e(const float* __restrict__ e,
    const float* __restrict__ Uew, const float* __restrict__ Ueb,
    float* __restrict__ ws, const float* __restrict__ bnew,
    const float* __restrict__ bneb, float* __restrict__ eout) {
  __shared__ float As[2][TILE_F];                      // double buffer, 33.8 KB
  const int blk  = blockIdx.x;                         // 2048 blocks = (b,i)
  const int b    = blk >> 8;
  const int i    = blk & 255;
  const int tid  = threadIdx.x;
  const int lane = tid & 31;
  const int wv   = tid >> 5;                           // wave 0..7 -> h-tile
  const int h0   = wv * 16;
  const int ln16 = lane & 15;
  const int lhi  = lane >> 4;                          // 0/1
  const int h    = h0 + ln16;

  const float* Vx  = ws + WS_VX;
  const float* Vnx = ws + WS_VNX;
  const float* erow0 = e + ((size_t)(b * N_ + i)) * N_ * H_;   // 256 x 128 slab

  // ---- B fragments: B[k][n] = Ue_w[h0+n][k]; 32 k-steps resident in VGPRs, reused 8x ----
  v2f bfrag[32];
  {
    const float* wrow = Uew + (h0 + ln16) * H_ + 2 * lhi;
#pragma unroll
    for (int kk = 0; kk < 32; ++kk)
      bfrag[kk] = *(const v2f*)(wrow + 4 * kk);
  }

  // bias + Vx_i folded into accumulator init (lane's h constant over all 8 acc regs)
  const float c0 = Ueb[h] + Vx[(b * N_ + i) * H_ + h];

  float me = 0.f, re = 0.f, gw = 0.f, gb = 0.f;
  if (PASS2) {
    me = ws[WS_STATS + h];
    re = ws[WS_STATS + 128 + h];
    gw = bnew[h];
    gb = bneb[h];
  }

  float s = 0.f, s2 = 0.f, ag = 0.f;

  // prime the pipeline: tiles 0 and 1 in flight
  issue_tile_async(erow0, tid, 0);
  issue_tile_async(erow0 + 32 * H_, tid, 1);

  for (int jt = 0; jt < 8; ++jt) {
    const int cur = jt & 1;
    const int j0  = jt * 32;
    // in-order per-wave async completion: <=4 outstanding => current tile landed
    if (jt < 7) asm volatile("s_wait_asynccnt 0x4" ::: "memory");
    else        asm volatile("s_wait_asynccnt 0x0" ::: "memory");
    __syncthreads();                                   // all waves' tile data visible

#pragma unroll
    for (int jsub = 0; jsub < 2; ++jsub) {
      v8f acc = {c0, c0, c0, c0, c0, c0, c0, c0};
      const float* arow = &As[cur][(jsub * 16 + ln16) * 132 + 2 * lhi];
#pragma unroll
      for (int kk = 0; kk < 32; ++kk) {
        v2f a = *(const v2f*)(arow + 4 * kk);
        acc = __builtin_amdgcn_wmma_f32_16x16x4_f32(
            false, a, false, bfrag[kk], (short)0, acc, false, false);
      }
#pragma unroll
      for (int r = 0; r < 8; ++r) {
        int jloc = jsub * 16 + r + 8 * lhi;            // row within tile
        int j    = j0 + jloc;
        float v  = acc[r] + Vx[(b * N_ + j) * H_ + h]; // e_new element
        if (PASS2) {
          float rel  = fmaxf((v - me) * re * gw + gb, 0.f);
          float eres = As[cur][jloc * 132 + h];        // original e from LDS
          eout[(((size_t)(b * N_ + i)) * N_ + j) * H_ + h] = eres + rel;
        } else {
          s  += v;
          s2 += v * v;
          float g = 1.f / (1.f + __expf(-v));          // sigmoid gate
          ag += g * Vnx[(b * N_ + j) * H_ + h];
        }
      }
    }

    __syncthreads();                                   // everyone done with buf `cur`
    if (jt + 2 < 8) issue_tile_async(erow0 + (j0 + 64) * H_, tid, cur);
  }

  if (!PASS2) {
    // lanes l and l+16 share the same h -> one xor-shuffle reduce
    s  += __shfl_xor(s, 16);
    s2 += __shfl_xor(s2, 16);
    ag += __shfl_xor(ag, 16);
    if (lhi == 0) {
      int slot = blk & 63;
      atomicAdd(ws + WS_EPART + slot * 256 + h, s);
      atomicAdd(ws + WS_EPART + slot * 256 + 128 + h, s2);
      ws[WS_AGG + (b * N_ + i) * H_ + h] = ag;         // full j-reduction done here: plain store
    }
  }
}

// x_new = x @ Un_w^T + Un_b + agg ; accumulate x BN partials
__global__ __launch_bounds__(256) void k_xnew(const float* __restrict__ x,
    const float* __restrict__ Unw, const float* __restrict__ Unb,
    float* __restrict__ ws) {
  int idx = blockIdx.x * 256 + threadIdx.x;            // 262144 total
  int row = idx >> 7, h = idx & 127;
  const float* xr = x + row * H_;
  const float* wr = Unw + h * H_;
  float a = 0.f;
#pragma unroll
  for (int k4 = 0; k4 < 32; ++k4) {
    float4 xv = *(const float4*)(xr + 4 * k4);
    float4 wv = *(const float4*)(wr + 4 * k4);
    a += xv.x * wv.x + xv.y * wv.y + xv.z * wv.z + xv.w * wv.w;
  }
  float v = a + Unb[h] + ws[WS_AGG + idx];
  ws[WS_XNEW + idx] = v;
  int slot = blockIdx.x & 63;
  atomicAdd(ws + WS_XPART + slot * 256 + h, v);
  atomicAdd(ws + WS_XPART + slot * 256 + 128 + h, v * v);
}

// finalize BN statistics for both e and x
__global__ void k_stats(float* __restrict__ ws) {
  int h = threadIdx.x;
  if (h >= 128) return;
  float se = 0.f, se2 = 0.f, sx = 0.f, sx2 = 0.f;
  for (int s = 0; s < 64; ++s) {
    se  += ws[WS_EPART + s * 256 + h];
    se2 += ws[WS_EPART + s * 256 + 128 + h];
    sx  += ws[WS_XPART + s * 256 + h];
    sx2 += ws[WS_XPART + s * 256 + 128 + h];
  }
  const float iMe = 1.0f / (float)(B_ * N_ * N_);
  const float iMx = 1.0f / (float)(B_ * N_);
  float mme = se * iMe, vve = se2 * iMe - mme * mme;
  float mmx = sx * iMx, vvx = sx2 * iMx - mmx * mmx;
  ws[WS_STATS + h]       = mme;
  ws[WS_STATS + 128 + h] = rsqrtf(vve + EPS_);
  ws[WS_STATS + 256 + h] = mmx;
  ws[WS_STATS + 384 + h] = rsqrtf(vvx + EPS_);
}

// x_out = x + relu(bn(x_new))
__global__ __launch_bounds__(256) void k_xout(const float* __restrict__ x,
    const float* __restrict__ bnnw, const float* __restrict__ bnnb,
    const float* __restrict__ ws, float* __restrict__ xout) {
  int idx = blockIdx.x * 256 + threadIdx.x;            // 262144 total
  int h = idx & 127;
  float v = ws[WS_XNEW + idx];
  float r = fmaxf((v - ws[WS_STATS + 256 + h]) * ws[WS_STATS + 384 + h] * bnnw[h] + bnnb[h], 0.f);
  xout[idx] = x[idx] + r;
}

extern "C" void kernel_launch(void* const* d_in, const int* in_sizes, int n_in,
                              void* d_out, int out_size, void* d_ws, size_t ws_size,
                              hipStream_t stream) {
  (void)in_sizes; (void)n_in; (void)out_size; (void)ws_size;
  const float* x    = (const float*)d_in[0];
  const float* e    = (const float*)d_in[1];
  const float* Uew  = (const float*)d_in[2];
  const float* Ueb  = (const float*)d_in[3];
  const float* Vew  = (const float*)d_in[4];
  const float* Veb  = (const float*)d_in[5];
  const float* Unw  = (const float*)d_in[6];
  const float* Unb  = (const float*)d_in[7];
  const float* Vnw  = (const float*)d_in[8];
  const float* Vnb  = (const float*)d_in[9];
  const float* bnew = (const float*)d_in[10];
  const float* bneb = (const float*)d_in[11];
  const float* bnnw = (const float*)d_in[12];
  const float* bnnb = (const float*)d_in[13];
  float* out  = (float*)d_out;
  float* ws   = (float*)d_ws;
  float* xout = out;                 // [8,256,128]
  float* eout = out + 262144;        // [8,256,256,128]

  k_init <<<512, 256, 0, stream>>>(ws);
  k_vxvnx<<<1024, 256, 0, stream>>>(x, Vew, Veb, Vnw, Vnb, ws);
  k_e<false><<<2048, 256, 0, stream>>>(e, Uew, Ueb, ws, nullptr, nullptr, nullptr);
  k_xnew <<<1024, 256, 0, stream>>>(x, Unw, Unb, ws);
  k_stats<<<1, 128, 0, stream>>>(ws);
  k_xout <<<1024, 256, 0, stream>>>(x, bnnw, bnnb, ws, xout);
  k_e<true><<<2048, 256, 0, stream>>>(e, Uew, Ueb, ws, bnew, bneb, eout);
}